// DepthToVoxelConverter_49306224558423
// MI455X (gfx1250) — compile-verified
//
#include <hip/hip_runtime.h>

// Depth→voxel scatter-reduce for MI455X (gfx1250, wave32).
// No matmul exists in this op → WMMA inapplicable; optimize the memory path:
//  - NT (non-temporal) streaming loads for the 128 MiB rgbd input
//  - no-return global_atomic_add_f32 into a 128 MiB accumulator that fits in
//    the 192 MB L2
//  - global_prefetch_b8 on the depth stream
//  - B128 vector loads (4 pixels/thread), predicated RGB loads (~25% valid)

typedef float f32x4 __attribute__((ext_vector_type(4)));

namespace {
constexpr int kB   = 32;
constexpr int kH   = 512;
constexpr int kW   = 512;
constexpr int kV   = 64;
constexpr int kV3  = kV * kV * kV;       // 262144 voxels per (b, channel) plane
constexpr int kHW  = kH * kW;            // 262144 pixels per (b, channel) plane
constexpr int kNPix4 = (kB * kHW) / 4;   // 2097152 threads, 4 px each
constexpr int kNVox4 = (kB * kV3) / 4;   // 2097152 threads, 4 voxels each
} // namespace

// ---------------------------------------------------------------------------
// Kernel 1: zero the accumulator (entire d_out). d_out/d_ws are poisoned and
// never re-poisoned between replays, so this must run every launch.
// ---------------------------------------------------------------------------
__global__ void __launch_bounds__(256) dv_zero(f32x4* __restrict__ out, int n4) {
  int i = blockIdx.x * blockDim.x + threadIdx.x;
  if (i < n4) out[i] = (f32x4)0.0f;
}

// ---------------------------------------------------------------------------
// Kernel 2: per-pixel unproject + voxel scatter with f32 atomics.
// Accumulator layout == final output layout: out[b][c][x][y][z],
// c=0: count (becomes occupancy), c=1..3: color sums (become means).
// ---------------------------------------------------------------------------
__global__ void __launch_bounds__(256) dv_scatter(const float* __restrict__ rgbd,
                                                  float* __restrict__ out) {
  int t = blockIdx.x * blockDim.x + threadIdx.x;
  if (t >= kNPix4) return;

  int pix0 = t << 2;              // 4 consecutive pixels along W (W % 4 == 0)
  int b    = pix0 / kHW;
  int rem  = pix0 - b * kHW;      // linear pixel index within the plane
  int h    = rem >> 9;            // / 512
  int w0   = rem & 511;           // % 512

  const float* rpl = rgbd + (b * 4 + 0) * kHW;
  const float* gpl = rgbd + (b * 4 + 1) * kHW;
  const float* bpl = rgbd + (b * 4 + 2) * kHW;
  const float* dpl = rgbd + (b * 4 + 3) * kHW;
  float*       ob  = out  + b * 4 * kV3;

  // CDNA5 speculative prefetch (global_prefetch_b8) ahead on the depth stream.
  if (rem + 8192 < kHW) __builtin_prefetch(dpl + rem + 8192, 0, 0);

  // Streaming B128 NT load of 4 depths.
  f32x4 d4 = __builtin_nontemporal_load((const f32x4*)(dpl + rem));

  // fx = fy = 256, cx = cy = 256; power-of-two scaling commutes exactly with
  // the reference's ((u-cx)*d)/fx order.
  const float yq = ((float)h - 256.0f) * (1.0f / 256.0f);

#pragma unroll
  for (int k = 0; k < 4; ++k) {
    float d = d4[k];
    float x = (((float)(w0 + k) - 256.0f) * (1.0f / 256.0f)) * d;
    float y = yq * d;
    // (p - SMIN)/(SMAX-SMIN)*(V-1) == (p + 2) * 15.75 exactly (15.75 = 63/4).
    // __float2int_rn == round-half-to-even == jnp.round.
    int ci = __float2int_rn((x + 2.0f) * 15.75f);
    int cj = __float2int_rn((y + 2.0f) * 15.75f);
    int ck = __float2int_rn((d + 2.0f) * 15.75f);
    // d<10 also rejects +inf; NaN fails both compares (covers isfinite()).
    bool ok = (d > 0.0f) & (d < 10.0f) &
              ((unsigned)ci < 64u) & ((unsigned)cj < 64u) & ((unsigned)ck < 64u);
    if (ok) {
      int lin = (ci * kV + cj) * kV + ck;
      int p   = rem + k;
      // No-return global_atomic_add_f32; target planes are L2-resident (128 MiB
      // accumulator < 192 MB L2).
      atomicAdd(ob + lin, 1.0f);
      atomicAdd(ob + kV3     + lin, __builtin_nontemporal_load(rpl + p));
      atomicAdd(ob + 2 * kV3 + lin, __builtin_nontemporal_load(gpl + p));
      atomicAdd(ob + 3 * kV3 + lin, __builtin_nontemporal_load(bpl + p));
    }
  }
}

// ---------------------------------------------------------------------------
// Kernel 3: in-place finalize. occ = (cnt > 0); color = cnt>0 ? sum/cnt : 0.
// (max(cnt,1)==cnt whenever cnt>0, so plain divide matches the reference.)
// ---------------------------------------------------------------------------
__global__ void __launch_bounds__(256) dv_finalize(float* __restrict__ out) {
  int t = blockIdx.x * blockDim.x + threadIdx.x;
  if (t >= kNVox4) return;

  int v0 = t << 2;                // 4 consecutive voxels (z-major), V3 % 4 == 0
  int b  = v0 / kV3;
  int r  = v0 - b * kV3;
  float* ob = out + b * 4 * kV3;

  f32x4 cnt = *(const f32x4*)(ob + r);
  f32x4 sr  = *(const f32x4*)(ob + kV3 + r);
  f32x4 sg  = *(const f32x4*)(ob + 2 * kV3 + r);
  f32x4 sb  = *(const f32x4*)(ob + 3 * kV3 + r);

  f32x4 occ, cr, cg, cb;
#pragma unroll
  for (int k = 0; k < 4; ++k) {
    bool nz = cnt[k] > 0.0f;
    float c = nz ? cnt[k] : 1.0f;
    occ[k] = nz ? 1.0f : 0.0f;
    cr[k]  = nz ? sr[k] / c : 0.0f;
    cg[k]  = nz ? sg[k] / c : 0.0f;
    cb[k]  = nz ? sb[k] / c : 0.0f;
  }

  *(f32x4*)(ob + r)           = occ;
  *(f32x4*)(ob + kV3 + r)     = cr;
  *(f32x4*)(ob + 2 * kV3 + r) = cg;
  *(f32x4*)(ob + 3 * kV3 + r) = cb;
}

// ---------------------------------------------------------------------------
extern "C" void kernel_launch(void* const* d_in, const int* in_sizes, int n_in,
                              void* d_out, int out_size, void* d_ws, size_t ws_size,
                              hipStream_t stream) {
  (void)in_sizes; (void)n_in; (void)d_ws; (void)ws_size;
  const float* rgbd = (const float*)d_in[0];
  float*       out  = (float*)d_out;

  int n4 = out_size / 4;  // 8388608 f32x4 chunks of the 128 MiB output
  dv_zero<<<(n4 + 255) / 256, 256, 0, stream>>>((f32x4*)out, n4);
  dv_scatter<<<(kNPix4 + 255) / 256, 256, 0, stream>>>(rgbd, out);
  dv_finalize<<<(kNVox4 + 255) / 256, 256, 0, stream>>>(out);
}